// GraphEncoder_81836306858308
// MI455X (gfx1250) — compile-verified
//
#include <hip/hip_runtime.h>
#include <math.h>

// GraphEncoder (2-layer GAT, H=2, D=128) for MI455X / gfx1250.
// fp32 end-to-end; GEMMs run on the matrix pipe via V_WMMA_F32_16X16X4_F32,
// with a distance-2 software pipeline (two operand buffers in flight) so no
// WMMA waits on a load issued in the same iteration.

typedef __attribute__((ext_vector_type(2))) float v2f;
typedef __attribute__((ext_vector_type(8))) float v8f;

#define DD 128      // per-head dim
#define HD 256      // H * D
#define SLOPE_GAT 0.2f
#define SLOPE_ACT 0.01f

// ---------------------------------------------------------------- utilities

__global__ void fill_f32(float* __restrict__ p, float v, long n) {
  long i = (long)blockIdx.x * blockDim.x + threadIdx.x;
  if (i < n) p[i] = v;
}

__global__ void gather_emb_k(const int* __restrict__ x, const float* __restrict__ emb,
                             float* __restrict__ h, int N) {
  long t = (long)blockIdx.x * blockDim.x + threadIdx.x;   // over N * (128/4)
  if (t >= (long)N * 32) return;
  int n = (int)(t >> 5);
  int c = (int)(t & 31);
  ((float4*)h)[(long)n * 32 + c] = ((const float4*)emb)[(long)x[n] * 32 + c];
}

// ------------------------------------------------------- fp32 WMMA GEMM
// C[M x Nout] = A[M x K] * B[K x Nout] (+ bias). One wave computes a 16x64
// slab of C with 4 accumulators. Distance-2 pipeline: while the 4 WMMAs of
// K-step k execute, the loads for step k+8 are already in flight and the
// loads for step k+4 are complete.
// f32 A 16x4 layout : lanes 0-15 -> K = k,k+1 ; lanes 16-31 -> K = k+2,k+3
// f32 B 4x16 layout : N = lane&15 ; same K split across lane halves
// f32 C/D layout    : VGPR j -> row (j + 8*(lane>>4)), col = lane&15
// Requires K % 8 == 0 (true here: K in {128, 256}).

#define LOAD_STEP(areg, breg)                       \
  do {                                              \
    (areg).x = Ap[0];                               \
    (areg).y = Ap[1];                               \
    (breg)[0].x = Bp[0];   (breg)[0].y = Bp[Nout];      \
    (breg)[1].x = Bp[16];  (breg)[1].y = Bp[Nout + 16]; \
    (breg)[2].x = Bp[32];  (breg)[2].y = Bp[Nout + 32]; \
    (breg)[3].x = Bp[48];  (breg)[3].y = Bp[Nout + 48]; \
    Ap += 4;                                        \
    Bp += bstep;                                    \
  } while (0)

#define WMMA4(areg, breg)                                                                                 \
  do {                                                                                                    \
    acc[0] = __builtin_amdgcn_wmma_f32_16x16x4_f32(false, (areg), false, (breg)[0], (short)0, acc[0], false, false); \
    acc[1] = __builtin_amdgcn_wmma_f32_16x16x4_f32(false, (areg), false, (breg)[1], (short)0, acc[1], false, false); \
    acc[2] = __builtin_amdgcn_wmma_f32_16x16x4_f32(false, (areg), false, (breg)[2], (short)0, acc[2], false, false); \
    acc[3] = __builtin_amdgcn_wmma_f32_16x16x4_f32(false, (areg), false, (breg)[3], (short)0, acc[3], false, false); \
  } while (0)

__global__ __launch_bounds__(128) void gemm_wmma_f32(
    const float* __restrict__ A, const float* __restrict__ B,
    const float* __restrict__ bias, float* __restrict__ C,
    int M, int K, int Nout) {
  const int lane   = threadIdx.x & 31;
  const int wv     = threadIdx.x >> 5;
  const int nSuper = Nout >> 6;                 // 64-wide supertiles
  const int total  = (M >> 4) * nSuper;
  const int tile   = blockIdx.x * 4 + wv;       // wave-uniform
  if (tile >= total) return;                    // whole-wave exit: EXEC uniform
  const int m0   = (tile / nSuper) << 4;
  const int n0   = (tile % nSuper) << 6;
  const int half = lane >> 4;                   // 0 or 1
  const int mr   = lane & 15;

  v8f acc[4] = {};

  // Incremented base pointers (no per-iteration 64-bit index rebuilds).
  const float* __restrict__ Ap = A + (long)(m0 + mr) * K + 2 * half;
  const float* __restrict__ Bp = B + (long)(2 * half) * Nout + n0 + mr;
  const long bstep = (long)4 * Nout;

  // ---- prologue: preload K-steps 0 and 4 into the two pipeline buffers
  v2f a0, a1, b0[4], b1[4];
  LOAD_STEP(a0, b0);          // step 0
  LOAD_STEP(a1, b1);          // step 4

  // ---- main loop: each trip consumes steps k and k+4, prefetches k+8, k+12
  int k = 0;
  for (; k + 16 <= K; k += 8) {
    WMMA4(a0, b0);            // step k      (loaded 1 trip ago)
    LOAD_STEP(a0, b0);        // step k+8
    WMMA4(a1, b1);            // step k+4
    LOAD_STEP(a1, b1);        // step k+12
  }

  // ---- epilogue: the last two preloaded steps
  WMMA4(a0, b0);
  WMMA4(a1, b1);

#pragma unroll
  for (int j = 0; j < 4; ++j) {
#pragma unroll
    for (int r = 0; r < 8; ++r) {
      const int row = m0 + r + 8 * half;
      const int col = n0 + j * 16 + mr;
      float v = acc[j][r];
      if (bias) v += bias[col];
      C[(long)row * Nout + col] = v;
    }
  }
}

// ------------------------------------------------------- attention logits
// al_s[n,h] = dot(xs[n,h,:], a_src[h,:])  (and a_dst)

__global__ void attn_logits(const float* __restrict__ xs,
                            const float* __restrict__ a_src,
                            const float* __restrict__ a_dst,
                            float* __restrict__ alS, float* __restrict__ alD, int N) {
  int t = blockIdx.x * blockDim.x + threadIdx.x;   // over N*2
  if (t >= N * 2) return;
  int n = t >> 1, h = t & 1;
  const float4* xr = (const float4*)(xs + (long)n * HD + h * DD);
  const float4* as = (const float4*)(a_src + h * DD);
  const float4* ad = (const float4*)(a_dst + h * DD);
  float ss = 0.f, sd = 0.f;
#pragma unroll 8
  for (int i = 0; i < 32; ++i) {
    float4 v = xr[i], s4 = as[i], d4 = ad[i];
    ss += v.x * s4.x + v.y * s4.y + v.z * s4.z + v.w * s4.w;
    sd += v.x * d4.x + v.y * d4.y + v.z * d4.z + v.w * d4.w;
  }
  alS[t] = ss; alD[t] = sd;
}

// ------------------------------------------------------- edge kernels

__device__ inline void edge_sd(const int* __restrict__ ei, int E, int e, int& s, int& d) {
  if (e < E) { s = ei[e]; d = ei[E + e]; }
  else       { s = e - E; d = e - E; }          // appended self-loops
}

__device__ inline void atomicMaxF(float* addr, float val) {
  if (val >= 0.f) atomicMax((int*)addr, __float_as_int(val));
  else            atomicMin((unsigned int*)addr, __float_as_uint(val));
}

__global__ void k_edge_alpha(const int* __restrict__ ei, int E, int N,
                             const float* __restrict__ alS, const float* __restrict__ alD,
                             float* __restrict__ alpha, float* __restrict__ mx) {
  int e = blockIdx.x * blockDim.x + threadIdx.x;
  if (e >= E + N) return;
  int s, d; edge_sd(ei, E, e, s, d);
#pragma unroll
  for (int h = 0; h < 2; ++h) {
    float a = alS[s * 2 + h] + alD[d * 2 + h];
    a = a > 0.f ? a : SLOPE_GAT * a;            // leaky_relu(0.2)
    alpha[(long)e * 2 + h] = a;
    atomicMaxF(&mx[d * 2 + h], a);
  }
}

__global__ void k_edge_exp(const int* __restrict__ ei, int E, int N,
                           const float* __restrict__ mx,
                           float* __restrict__ alpha,   // in: logit, out: exp
                           float* __restrict__ sm) {
  int e = blockIdx.x * blockDim.x + threadIdx.x;
  if (e >= E + N) return;
  int s, d; edge_sd(ei, E, e, s, d);
#pragma unroll
  for (int h = 0; h < 2; ++h) {
    float ea = __expf(alpha[(long)e * 2 + h] - mx[d * 2 + h]);
    alpha[(long)e * 2 + h] = ea;
    atomicAdd(&sm[d * 2 + h], ea);
  }
}

// One wave per edge: 32 lanes x 8 cols = 256 message values.
// lanes 0-15 -> head 0, lanes 16-31 -> head 1.
__global__ void k_edge_agg(const int* __restrict__ ei, int E, int N,
                           const float* __restrict__ xs, const float* __restrict__ ea,
                           const float* __restrict__ sm, float* __restrict__ out) {
  long t = (long)blockIdx.x * blockDim.x + threadIdx.x;
  int e = (int)(t >> 5);
  int lane = (int)(t & 31);
  if (e >= E + N) return;
  int s, d; edge_sd(ei, E, e, s, d);
  const int head = lane >> 4;
  const float w = ea[(long)e * 2 + head] / sm[d * 2 + head];
  const float* __restrict__ xr = xs + (long)s * HD + lane * 8;
  float* __restrict__ orow = out + (long)d * HD + lane * 8;
  float4 x0 = *(const float4*)xr;
  float4 x1 = *(const float4*)(xr + 4);
  atomicAdd(orow + 0, x0.x * w);
  atomicAdd(orow + 1, x0.y * w);
  atomicAdd(orow + 2, x0.z * w);
  atomicAdd(orow + 3, x0.w * w);
  atomicAdd(orow + 4, x1.x * w);
  atomicAdd(orow + 5, x1.y * w);
  atomicAdd(orow + 6, x1.z * w);
  atomicAdd(orow + 7, x1.w * w);
}

// ------------------------------------------------------- bias + leaky(0.01)

__global__ void bias_act_k(const float* __restrict__ in, const float* __restrict__ b,
                           float* __restrict__ out, long total) {
  long i = (long)blockIdx.x * blockDim.x + threadIdx.x;
  if (i >= total) return;
  int c = (int)(i & (HD - 1));
  float v = in[i] + b[c];
  out[i] = v > 0.f ? v : SLOPE_ACT * v;
}

// ------------------------------------------------------- mean pooling

__global__ void pool_count(const int* __restrict__ batch, int N, float* __restrict__ cnt) {
  int n = blockIdx.x * blockDim.x + threadIdx.x;
  if (n < N) atomicAdd(&cnt[batch[n]], 1.0f);
}

__global__ void pool_sum(const float* __restrict__ h, const int* __restrict__ batch,
                         int N, float* __restrict__ z) {
  long t = (long)blockIdx.x * blockDim.x + threadIdx.x;   // over N*128
  if (t >= (long)N * DD) return;
  int n = (int)(t >> 7), c = (int)(t & 127);
  atomicAdd(&z[(long)batch[n] * DD + c], h[(long)n * DD + c]);
}

__global__ void pool_div(float* __restrict__ z, const float* __restrict__ cnt, int G) {
  int t = blockIdx.x * blockDim.x + threadIdx.x;          // over G*128
  if (t >= G * DD) return;
  z[t] /= fmaxf(cnt[t >> 7], 1.0f);
}

// ------------------------------------------------------- host orchestration

static inline int cdiv_i(long a, int b) { return (int)((a + b - 1) / b); }

static void launch_gemm(const float* A, const float* B, const float* bias, float* C,
                        int M, int K, int Nout, hipStream_t st) {
  int tiles = (M / 16) * (Nout / 64);
  gemm_wmma_f32<<<cdiv_i(tiles, 4), 128, 0, st>>>(A, B, bias, C, M, K, Nout);
}

static void gat_layer(const float* Ain, int K, const float* W,
                      const float* a_src, const float* a_dst,
                      const int* ei, int E, int N,
                      float* xs, float* alS, float* alD, float* mx, float* sm,
                      float* alpha, float* out, hipStream_t st) {
  const int ET = E + N;
  launch_gemm(Ain, W, nullptr, xs, N, K, HD, st);
  attn_logits<<<cdiv_i((long)N * 2, 256), 256, 0, st>>>(xs, a_src, a_dst, alS, alD, N);
  fill_f32<<<cdiv_i((long)N * 2, 256), 256, 0, st>>>(mx, -3.0e38f, (long)N * 2);
  fill_f32<<<cdiv_i((long)N * 2, 256), 256, 0, st>>>(sm, 0.f, (long)N * 2);
  fill_f32<<<cdiv_i((long)N * HD, 256), 256, 0, st>>>(out, 0.f, (long)N * HD);
  k_edge_alpha<<<cdiv_i(ET, 256), 256, 0, st>>>(ei, E, N, alS, alD, alpha, mx);
  k_edge_exp<<<cdiv_i(ET, 256), 256, 0, st>>>(ei, E, N, mx, alpha, sm);
  k_edge_agg<<<cdiv_i((long)ET * 32, 256), 256, 0, st>>>(ei, E, N, xs, alpha, sm, out);
}

extern "C" void kernel_launch(void* const* d_in, const int* in_sizes, int n_in,
                              void* d_out, int out_size, void* d_ws, size_t ws_size,
                              hipStream_t stream) {
  const int*   x      = (const int*)d_in[0];
  const int*   ei     = (const int*)d_in[1];
  const int*   batch  = (const int*)d_in[2];
  const float* emb    = (const float*)d_in[3];
  const float* W1     = (const float*)d_in[4];
  const float* a_src1 = (const float*)d_in[5];
  const float* a_dst1 = (const float*)d_in[6];
  const float* b1     = (const float*)d_in[7];
  const float* W2     = (const float*)d_in[8];
  const float* a_src2 = (const float*)d_in[9];
  const float* a_dst2 = (const float*)d_in[10];
  const float* b2     = (const float*)d_in[11];
  const float* Wp     = (const float*)d_in[12];
  const float* bp     = (const float*)d_in[13];

  const int N  = in_sizes[0];
  const int E  = in_sizes[1] / 2;
  const int ET = E + N;
  const int G  = (int)(((long)out_size - (long)N * DD) / DD);

  // ---- workspace layout (floats) ----
  float* ws     = (float*)d_ws;
  float* bufA   = ws;                              // N*256  (layer input)
  float* bufXS  = bufA  + (long)N * HD;            // N*256  (xs = A @ W)
  float* bufOut = bufXS + (long)N * HD;            // N*256  (aggregated)
  float* alS    = bufOut + (long)N * HD;           // N*2
  float* alD    = alS + (long)N * 2;               // N*2
  float* mx     = alD + (long)N * 2;               // N*2
  float* sm     = mx  + (long)N * 2;               // N*2
  float* alpha  = sm  + (long)N * 2;               // ET*2
  float* cnt    = alpha + (long)ET * 2;            // G

  float* h_out = (float*)d_out;                    // N*128
  float* z_out = h_out + (long)N * DD;             // G*128

  // 1) h0 = emb[x]  (dense N x 128 into bufA)
  gather_emb_k<<<cdiv_i((long)N * 32, 256), 256, 0, stream>>>(x, emb, bufA, N);

  // 2) GAT layer 1: bufA(N x 128) -> bufOut(N x 256)
  gat_layer(bufA, DD, W1, a_src1, a_dst1, ei, E, N,
            bufXS, alS, alD, mx, sm, alpha, bufOut, stream);
  bias_act_k<<<cdiv_i((long)N * HD, 256), 256, 0, stream>>>(bufOut, b1, bufA, (long)N * HD);

  // 3) GAT layer 2: bufA(N x 256) -> bufOut(N x 256)
  gat_layer(bufA, HD, W2, a_src2, a_dst2, ei, E, N,
            bufXS, alS, alD, mx, sm, alpha, bufOut, stream);
  bias_act_k<<<cdiv_i((long)N * HD, 256), 256, 0, stream>>>(bufOut, b2, bufA, (long)N * HD);

  // 4) projection: h = bufA @ Wp + bp  -> d_out (N x 128)
  launch_gemm(bufA, Wp, bp, h_out, N, HD, DD, stream);

  // 5) mean pooling per graph -> z (G x 128)
  fill_f32<<<cdiv_i((long)G * DD, 256), 256, 0, stream>>>(z_out, 0.f, (long)G * DD);
  fill_f32<<<cdiv_i(G, 256), 256, 0, stream>>>(cnt, 0.f, (long)G);
  pool_count<<<cdiv_i(N, 256), 256, 0, stream>>>(batch, N, cnt);
  pool_sum<<<cdiv_i((long)N * DD, 256), 256, 0, stream>>>(h_out, batch, N, z_out);
  pool_div<<<cdiv_i((long)G * DD, 256), 256, 0, stream>>>(z_out, cnt, G);
}